// WeaveLayer_20478404067942
// MI455X (gfx1250) — compile-verified
//
#include <hip/hip_runtime.h>

typedef float v2f __attribute__((ext_vector_type(2)));
typedef float v8f __attribute__((ext_vector_type(8)));

#define BB 64
#define NN 128
#define FA 75
#define FP 14
#define HH 50

// workspace layout (float offsets)
#define OFF_U1 0            // [75][50]  W_AA @ W_A[:50]
#define OFF_U2 3750         // [14][50]  W_PA @ W_A[50:]
#define OFF_V  4450         // [75][50]  (W_AP[:75]+W_AP[75:]) @ W_P[:50]
#define OFF_WF 8200         // [16][64]  padded: W_PP @ W_P[50:]
#define OFF_EA 9224         // [50]      fused A bias
#define OFF_DP 9274         // [50]      fused P bias
#define OFF_Q  9324         // [B*N][50] broadcast vector q

// ---------------------------------------------------------------------------
// Kernel 1: fuse all the small weight matrices on-device (one block).
// ---------------------------------------------------------------------------
__global__ void fuse_weights(const float* __restrict__ W_AA, const float* __restrict__ b_AA,
                             const float* __restrict__ W_PA, const float* __restrict__ b_PA,
                             const float* __restrict__ W_A,  const float* __restrict__ b_A,
                             const float* __restrict__ W_AP, const float* __restrict__ b_AP,
                             const float* __restrict__ W_PP, const float* __restrict__ b_PP,
                             const float* __restrict__ W_P,  const float* __restrict__ b_P,
                             float* __restrict__ ws) {
    const int t = threadIdx.x;
    const int nthr = blockDim.x;
    float* U1 = ws + OFF_U1;
    float* U2 = ws + OFF_U2;
    float* V  = ws + OFF_V;
    float* WF = ws + OFF_WF;
    float* EA = ws + OFF_EA;
    float* DP = ws + OFF_DP;

    // U1[f][h] = sum_k W_AA[f][k] * W_A[k][h]
    for (int idx = t; idx < FA * HH; idx += nthr) {
        int f = idx / HH, h = idx % HH;
        float s = 0.f;
        for (int k = 0; k < HH; ++k) s += W_AA[f * HH + k] * W_A[k * HH + h];
        U1[idx] = s;
    }
    // U2[f][h] = sum_k W_PA[f][k] * W_A[50+k][h]
    for (int idx = t; idx < FP * HH; idx += nthr) {
        int f = idx / HH, h = idx % HH;
        float s = 0.f;
        for (int k = 0; k < HH; ++k) s += W_PA[f * HH + k] * W_A[(HH + k) * HH + h];
        U2[idx] = s;
    }
    // V[f][h] = sum_k (W_AP[f][k] + W_AP[75+f][k]) * W_P[k][h]
    for (int idx = t; idx < FA * HH; idx += nthr) {
        int f = idx / HH, h = idx % HH;
        float s = 0.f;
        for (int k = 0; k < HH; ++k)
            s += (W_AP[f * HH + k] + W_AP[(FA + f) * HH + k]) * W_P[k * HH + h];
        V[idx] = s;
    }
    // WF padded [16][64]: zero then fill Wf[f][h] = sum_k W_PP[f][k] * W_P[50+k][h]
    for (int idx = t; idx < 16 * 64; idx += nthr) WF[idx] = 0.f;
    __syncthreads();
    for (int idx = t; idx < FP * HH; idx += nthr) {
        int f = idx / HH, h = idx % HH;
        float s = 0.f;
        for (int k = 0; k < HH; ++k) s += W_PP[f * HH + k] * W_P[(HH + k) * HH + h];
        WF[f * 64 + h] = s;
    }
    // fused biases
    for (int h = t; h < HH; h += nthr) {
        float s1 = b_A[h], s2 = 0.f;
        for (int k = 0; k < HH; ++k) {
            s1 += b_AA[k] * W_A[k * HH + h];
            s2 += b_PA[k] * W_A[(HH + k) * HH + h];
        }
        EA[h] = s1 + (float)NN * s2;
        float d = b_P[h];
        for (int k = 0; k < HH; ++k)
            d += b_AP[k] * W_P[k * HH + h] + b_PP[k] * W_P[(HH + k) * HH + h];
        DP[h] = d;
    }
}

// ---------------------------------------------------------------------------
// Kernel 2: per (b,i): pairsum over j, A output, and q vector. 64 threads.
// ---------------------------------------------------------------------------
__global__ __launch_bounds__(64) void atom_kernel(const float* __restrict__ atom,
                                                  const float* __restrict__ pair,
                                                  const float* __restrict__ amask,
                                                  const float* __restrict__ ws,
                                                  float* __restrict__ q,
                                                  float* __restrict__ A_out) {
    const int bi = blockIdx.x;           // b*N + i
    const int t  = threadIdx.x;          // 0..63
    __shared__ float sAtom[FA];
    __shared__ float sPart[64][FP];
    __shared__ float sSum[FP];

    const float* prow = pair + (size_t)bi * NN * FP;
    float ps[FP];
#pragma unroll
    for (int f = 0; f < FP; ++f) ps[f] = 0.f;
    for (int j = t; j < NN; j += 64) {
#pragma unroll
        for (int f = 0; f < FP; ++f) ps[f] += prow[j * FP + f];
    }
#pragma unroll
    for (int f = 0; f < FP; ++f) sPart[t][f] = ps[f];
    for (int f = t; f < FA; f += 64) sAtom[f] = atom[(size_t)bi * FA + f];
    __syncthreads();
    if (t < FP) {
        float s = 0.f;
        for (int w = 0; w < 64; ++w) s += sPart[w][t];
        sSum[t] = s;
    }
    __syncthreads();
    if (t < HH) {
        const float* U1 = ws + OFF_U1;
        const float* U2 = ws + OFF_U2;
        const float* V  = ws + OFF_V;
        const float* EA = ws + OFF_EA;
        float aacc = EA[t], qacc = 0.f;
        for (int f = 0; f < FA; ++f) {
            float av = sAtom[f];
            aacc += av * U1[f * HH + t];
            qacc += av * V[f * HH + t];
        }
#pragma unroll
        for (int f = 0; f < FP; ++f) aacc += sSum[f] * U2[f * HH + t];
        q[(size_t)bi * HH + t] = qacc;
        A_out[(size_t)bi * HH + t] = aacc * amask[bi];
    }
}

// ---------------------------------------------------------------------------
// Kernel 3: P[b,i,j,h] = (pair[b,i,j,:] @ Wf)[h] + q_i[h] + q_j[h] + dP[h],
// masked. WMMA f32 16x16x4. One block per (b,i); 8 waves, 1 M-tile each.
// ---------------------------------------------------------------------------
__global__ __launch_bounds__(256) void pair_kernel(const float* __restrict__ pair,
                                                   const float* __restrict__ pmask,
                                                   const float* __restrict__ ws,
                                                   const float* __restrict__ q,
                                                   float* __restrict__ P) {
    const int bi = blockIdx.x;           // b*N + i
    const int t  = threadIdx.x;
    __shared__ float sB[16 * 64];        // Wf padded, [k][n]
    __shared__ float sA[NN * 16];        // pair tile, [j][k] with K padded to 16

    // stage pair tile (coalesced: 1792 contiguous floats)
    const float* src = pair + (size_t)bi * NN * FP;
    for (int idx = t; idx < NN * FP; idx += 256) {
        int j = idx / FP, k = idx % FP;
        sA[j * 16 + k] = src[idx];
    }
    { // zero K padding (k = 14,15), 256 elements -> one step
        int j = t >> 1, k = FP + (t & 1);
        sA[j * 16 + k] = 0.f;
    }
    const float* WF = ws + OFF_WF;
    for (int idx = t; idx < 16 * 64; idx += 256) sB[idx] = WF[idx];
    __syncthreads();

    const int lane = t & 31;
    const int wave = t >> 5;
    const int hi = lane >> 4;            // 0 or 1
    const int lo = lane & 15;
    const int jbase = wave * 16;

    // B fragments: 4 n-tiles x 4 k-steps. Lane holds B[k0][n], B[k0+1][n].
    v2f bfrag[4][4];
#pragma unroll
    for (int nt = 0; nt < 4; ++nt)
#pragma unroll
        for (int kk = 0; kk < 4; ++kk) {
            int k0 = kk * 4 + 2 * hi;
            int n  = nt * 16 + lo;
            bfrag[nt][kk].x = sB[k0 * 64 + n];
            bfrag[nt][kk].y = sB[(k0 + 1) * 64 + n];
        }

    v8f acc[4] = {};
#pragma unroll
    for (int kk = 0; kk < 4; ++kk) {
        int k0 = kk * 4 + 2 * hi;
        v2f a;
        a.x = sA[(jbase + lo) * 16 + k0];
        a.y = sA[(jbase + lo) * 16 + k0 + 1];
#pragma unroll
        for (int nt = 0; nt < 4; ++nt) {
            acc[nt] = __builtin_amdgcn_wmma_f32_16x16x4_f32(
                false, a, false, bfrag[nt][kk], (short)0, acc[nt], false, false);
        }
    }

    // epilogue: + q_i[h] + q_j[h] + dP[h], * mask; NT stores (P >> L2, write-once)
    const float* DP = ws + OFF_DP;
    const int b = bi / NN;
#pragma unroll
    for (int nt = 0; nt < 4; ++nt) {
        int h = nt * 16 + lo;
        if (h < HH) {
            float base = q[(size_t)bi * HH + h] + DP[h];
            const float* accp = (const float*)&acc[nt];
#pragma unroll
            for (int v = 0; v < 8; ++v) {
                int j = jbase + v + 8 * hi;
                float m  = pmask[(size_t)bi * NN + j];
                float qj = q[((size_t)(b * NN + j)) * HH + h];
                float val = (accp[v] + base + qj) * m;
                __builtin_nontemporal_store(val, &P[((size_t)bi * NN + j) * HH + h]);
            }
        }
    }
}

// ---------------------------------------------------------------------------
extern "C" void kernel_launch(void* const* d_in, const int* in_sizes, int n_in,
                              void* d_out, int out_size, void* d_ws, size_t ws_size,
                              hipStream_t stream) {
    const float* atom  = (const float*)d_in[0];
    const float* pair  = (const float*)d_in[1];
    const float* amask = (const float*)d_in[2];
    const float* pmask = (const float*)d_in[3];
    const float* W_AA  = (const float*)d_in[4];
    const float* b_AA  = (const float*)d_in[5];
    const float* W_PA  = (const float*)d_in[6];
    const float* b_PA  = (const float*)d_in[7];
    const float* W_A   = (const float*)d_in[8];
    const float* b_A   = (const float*)d_in[9];
    const float* W_AP  = (const float*)d_in[10];
    const float* b_AP  = (const float*)d_in[11];
    const float* W_PP  = (const float*)d_in[12];
    const float* b_PP  = (const float*)d_in[13];
    const float* W_P   = (const float*)d_in[14];
    const float* b_P   = (const float*)d_in[15];

    float* ws    = (float*)d_ws;
    float* A_out = (float*)d_out;                              // [B,N,50]
    float* P_out = (float*)d_out + (size_t)BB * NN * HH;       // [B,N,N,50]
    float* q     = ws + OFF_Q;

    fuse_weights<<<1, 256, 0, stream>>>(W_AA, b_AA, W_PA, b_PA, W_A, b_A,
                                        W_AP, b_AP, W_PP, b_PP, W_P, b_P, ws);
    atom_kernel<<<BB * NN, 64, 0, stream>>>(atom, pair, amask, ws, q, A_out);
    pair_kernel<<<BB * NN, 256, 0, stream>>>(pair, pmask, ws, q, P_out);
}